// GCNEncoder_48172353192285
// MI455X (gfx1250) — compile-verified
//
#include <hip/hip_runtime.h>

typedef __attribute__((ext_vector_type(2))) float v2f;
typedef __attribute__((ext_vector_type(8))) float v8f;

// ---------------------------------------------------------------------------
// Degree / normalization kernels
// ---------------------------------------------------------------------------
__global__ void k_init_deg(float* __restrict__ deg, int n) {
  int i = blockIdx.x * blockDim.x + threadIdx.x;
  if (i < n) deg[i] = 1.0f;  // self-loop contributes 1
}

__global__ void k_count_deg(const long long* __restrict__ dst,
                            float* __restrict__ deg, long long ne) {
  long long i = (long long)blockIdx.x * blockDim.x + threadIdx.x;
  if (i < ne) unsafeAtomicAdd(&deg[(int)dst[i]], 1.0f);
}

__global__ void k_dinv(float* __restrict__ deg, int n) {
  int i = blockIdx.x * blockDim.x + threadIdx.x;
  if (i < n) {
    float d = deg[i];
    deg[i] = (d > 0.0f) ? rsqrtf(d) : 0.0f;
  }
}

// ---------------------------------------------------------------------------
// f32 WMMA GEMM: C[M x N] = A[M x K] @ B[K x N], row-major.
// One wave (32 threads) computes one 16x16 C tile via V_WMMA_F32_16X16X4_F32.
// ISA 16x4 f32 A layout: lanes 0-15 -> M=0..15 with K={k0,k0+1} in v[0],v[1];
// lanes 16-31 -> same M rows with K={k0+2,k0+3}. B (4x16) mirrors with N
// striped across lanes. C/D: VGPR r holds M=r (lanes 0-15) / M=8+r (16-31).
// ---------------------------------------------------------------------------
__global__ void k_gemm_wmma_f32(const float* __restrict__ A,
                                const float* __restrict__ B,
                                float* __restrict__ C,
                                int M, int N, int K) {
  const int lane = threadIdx.x & 31;
  const int half = lane >> 4;    // 0: K+{0,1}, 1: K+{2,3}
  const int ml   = lane & 15;    // M index for A frag, N index for B frag
  const int kb   = half * 2;
  const int m0 = blockIdx.x * 16;
  const int n0 = blockIdx.y * 16;

  const int ra = min(m0 + ml, M - 1);          // clamped row for A loads
  const float* __restrict__ Arow = A + (long long)ra * K;
  const float* __restrict__ Bcol = B + (n0 + ml);

  v8f acc = {0.f, 0.f, 0.f, 0.f, 0.f, 0.f, 0.f, 0.f};

  for (int k0 = 0; k0 < K; k0 += 4) {
    v2f a, b;
    a[0] = Arow[k0 + kb];
    a[1] = Arow[k0 + kb + 1];
    b[0] = Bcol[(long long)(k0 + kb) * N];
    b[1] = Bcol[(long long)(k0 + kb + 1) * N];
    acc = __builtin_amdgcn_wmma_f32_16x16x4_f32(
        /*neg_a=*/false, a, /*neg_b=*/false, b,
        /*c_mod=*/(short)0, acc, /*reuse_a=*/false, /*reuse_b=*/false);
  }

#pragma unroll
  for (int r = 0; r < 8; ++r) {
    int row = m0 + half * 8 + r;
    if (row < M) C[(long long)row * N + n0 + ml] = acc[r];
  }
}

// ---------------------------------------------------------------------------
// Aggregation: out[i][f] = bias[f] + dinv[i]^2 * h[i][f]   (self-loop + bias)
// ---------------------------------------------------------------------------
__global__ void k_agg_init(const float* __restrict__ h,
                           const float* __restrict__ dinv,
                           const float* __restrict__ bias,
                           float* __restrict__ out, int n, int nf) {
  int i = blockIdx.x * blockDim.x + threadIdx.x;
  if (i < n * nf) {
    int node = i / nf;
    int f = i - node * nf;
    float d = dinv[node];
    out[i] = bias[f] + d * d * h[i];
  }
}

// Per-edge scatter: out[dst] += dinv[src]*dinv[dst] * h[src]  (float4 chunks)
__global__ void k_agg_edges(const long long* __restrict__ src,
                            const long long* __restrict__ dst,
                            const float* __restrict__ dinv,
                            const float* __restrict__ h,
                            float* __restrict__ out,
                            long long ne, int nf) {
  const int tpe = nf >> 2;  // threads (float4 chunks) per edge
  long long t = (long long)blockIdx.x * blockDim.x + threadIdx.x;
  long long e = t / tpe;
  int c = (int)(t - e * tpe);
  if (e < ne) {
    int s = (int)src[e];
    int d = (int)dst[e];
    float w = dinv[s] * dinv[d];
    float4 v = *((const float4*)(h + (long long)s * nf) + c);
    float* o = out + (long long)d * nf + c * 4;
    unsafeAtomicAdd(o + 0, w * v.x);
    unsafeAtomicAdd(o + 1, w * v.y);
    unsafeAtomicAdd(o + 2, w * v.z);
    unsafeAtomicAdd(o + 3, w * v.w);
  }
}

__global__ void k_relu(float* __restrict__ x, int n) {
  int i = blockIdx.x * blockDim.x + threadIdx.x;
  if (i < n) x[i] = fmaxf(x[i], 0.0f);
}

// ---------------------------------------------------------------------------
// Launch
// ---------------------------------------------------------------------------
extern "C" void kernel_launch(void* const* d_in, const int* in_sizes, int n_in,
                              void* d_out, int out_size, void* d_ws, size_t ws_size,
                              hipStream_t stream) {
  const float*     x   = (const float*)d_in[0];
  const long long* ei  = (const long long*)d_in[1];  // int64 edge_index [2, E]
  const float*     W1  = (const float*)d_in[2];
  const float*     b1  = (const float*)d_in[3];
  const float*     W2  = (const float*)d_in[4];
  const float*     b2  = (const float*)d_in[5];
  float*           out = (float*)d_out;

  const int HID = in_sizes[3];              // 64
  const int OUT = in_sizes[5];              // 32
  const int IN  = in_sizes[2] / HID;        // 256
  const int N   = in_sizes[0] / IN;         // 100000
  const long long E = in_sizes[1] / 2;      // 3200000
  const long long* src = ei;
  const long long* dstp = ei + E;

  // Workspace carve-out (~52 MB): dinv | h (gemm out, reused) | agg
  char* ws = (char*)d_ws;
  size_t off = 0;
  auto carve = [&](size_t bytes) -> void* {
    void* p = ws + off;
    off = (off + bytes + 255) & ~(size_t)255;
    return p;
  };
  float* dinv = (float*)carve((size_t)N * sizeof(float));
  float* h    = (float*)carve((size_t)N * HID * sizeof(float));
  float* agg  = (float*)carve((size_t)N * HID * sizeof(float));

  const int T = 256;

  // ---- normalization: deg -> dinv ----
  k_init_deg<<<(N + T - 1) / T, T, 0, stream>>>(dinv, N);
  k_count_deg<<<(int)((E + T - 1) / T), T, 0, stream>>>(dstp, dinv, E);
  k_dinv<<<(N + T - 1) / T, T, 0, stream>>>(dinv, N);

  // ---- layer 1: h = x @ W1 ; agg = Â h + b1 ; relu ----
  {
    dim3 grid((N + 15) / 16, HID / 16);
    k_gemm_wmma_f32<<<grid, 32, 0, stream>>>(x, W1, h, N, HID, IN);
  }
  k_agg_init<<<(N * HID + T - 1) / T, T, 0, stream>>>(h, dinv, b1, agg, N, HID);
  {
    long long tot = E * (HID / 4);
    k_agg_edges<<<(int)((tot + T - 1) / T), T, 0, stream>>>(src, dstp, dinv, h, agg, E, HID);
  }
  k_relu<<<(N * HID + T - 1) / T, T, 0, stream>>>(agg, N * HID);

  // ---- layer 2: h2 = agg @ W2 (reuse h buffer) ; out = Â h2 + b2 ----
  {
    dim3 grid((N + 15) / 16, OUT / 16);
    k_gemm_wmma_f32<<<grid, 32, 0, stream>>>(agg, W2, h, N, OUT, HID);
  }
  k_agg_init<<<(N * OUT + T - 1) / T, T, 0, stream>>>(h, dinv, b2, out, N, OUT);
  {
    long long tot = E * (OUT / 4);
    k_agg_edges<<<(int)((tot + T - 1) / T), T, 0, stream>>>(src, dstp, dinv, h, out, E, OUT);
  }
}